// SpatialConv_90520730730507
// MI455X (gfx1250) — compile-verified
//
#include <hip/hip_runtime.h>
#include <hip/hip_bf16.h>

#define N_NODES 100000
#define N_EDGES 1600000
#define D 64
#define BN_EPS 1e-5f

typedef __attribute__((ext_vector_type(2))) float v2f;
typedef __attribute__((ext_vector_type(8))) float v8f;

// ---------------------------------------------------------------------------
// 1) init: deg = 1.0 (self loop), BN accumulators = 0
// ---------------------------------------------------------------------------
__global__ void init_kernel(float* __restrict__ deg, float* __restrict__ bnacc) {
    int i = blockIdx.x * blockDim.x + threadIdx.x;
    if (i < N_NODES) deg[i] = 1.0f;
    if (i < 2 * D) bnacc[i] = 0.0f;
}

// ---------------------------------------------------------------------------
// 2) degree: atomic count of edges per target node (L2-resident atomics)
// ---------------------------------------------------------------------------
__global__ void degree_kernel(const long long* __restrict__ ei, float* __restrict__ deg) {
    int e = blockIdx.x * blockDim.x + threadIdx.x;
    if (e >= N_EDGES) return;
    int t = (int)ei[N_EDGES + e];
    atomicAdd(&deg[t], 1.0f);
}

// ---------------------------------------------------------------------------
// 3) dinv = rsqrt(deg)
// ---------------------------------------------------------------------------
__global__ void rsqrt_kernel(const float* __restrict__ deg, float* __restrict__ dinv) {
    int i = blockIdx.x * blockDim.x + threadIdx.x;
    if (i < N_NODES) dinv[i] = rsqrtf(deg[i]);
}

// ---------------------------------------------------------------------------
// 4) h = x @ W^T via V_WMMA_F32_16X16X4_F32.
//    One wave -> one 16-node tile x all 64 output channels (4 WMMA tiles).
//    A layout: lane = M(0..15) + K-half select in bit4; VGPRs sweep K.
//    B layout: lane = N(0..15) + K-half select in bit4; VGPRs sweep K.
//    B[k][n] = W[n][k] (W is [out,in] row-major).
// ---------------------------------------------------------------------------
__global__ void gemm_wmma_kernel(const float* __restrict__ x,
                                 const float* __restrict__ W,
                                 float* __restrict__ h) {
    int wave = (blockIdx.x * blockDim.x + threadIdx.x) >> 5;  // global wave id
    int lane = threadIdx.x & 31;
    if (wave >= N_NODES / 16) return;  // wave-uniform: EXEC all-ones inside

    const int row   = wave * 16 + (lane & 15);
    const int khalf = (lane >> 4) * 2;     // K offset: 0 or 2
    const int col   = lane & 15;
    const float* __restrict__ xrow = x + row * D;

    const int mbase = (lane >> 4) * 8;
#pragma unroll
    for (int nt = 0; nt < 4; ++nt) {
        v8f acc = {};
        const int n = nt * 16 + col;
        const float* __restrict__ wrow = W + n * D;  // B[k][n] = W[n][k]
#pragma unroll
        for (int ks = 0; ks < 16; ++ks) {
            const int k = ks * 4 + khalf;
            v2f a, b;
            a.x = xrow[k];     a.y = xrow[k + 1];
            b.x = wrow[k];     b.y = wrow[k + 1];
            acc = __builtin_amdgcn_wmma_f32_16x16x4_f32(
                /*neg_a=*/false, a, /*neg_b=*/false, b,
                /*c_mod=*/(short)0, acc, /*reuse_a=*/false, /*reuse_b=*/false);
        }
#pragma unroll
        for (int v = 0; v < 8; ++v) {
            const int m = mbase + v;  // D layout: lanes 0-15 -> M=v, 16-31 -> M=v+8
            h[(wave * 16 + m) * D + nt * 16 + col] = acc[v];
        }
    }
}

// ---------------------------------------------------------------------------
// 5) seed out = h * (1/deg) + b   (self-loop term + bias; replaces zero pass)
//    float4 vectorized: 1.6M float4 elements.
// ---------------------------------------------------------------------------
__global__ void seed_out_kernel(const float* __restrict__ h,
                                const float* __restrict__ dinv,
                                const float* __restrict__ b,
                                float* __restrict__ out) {
    int i = blockIdx.x * blockDim.x + threadIdx.x;          // float4 index
    if (i >= (N_NODES * D) / 4) return;
    int row = i >> 4;                                       // 16 float4 per row
    int c4  = i & 15;
    float di = dinv[row];
    float invdeg = di * di;
    float4 hv = reinterpret_cast<const float4*>(h)[i];
    float4 bv = reinterpret_cast<const float4*>(b)[c4];
    float4 o;
    o.x = hv.x * invdeg + bv.x;
    o.y = hv.y * invdeg + bv.y;
    o.z = hv.z * invdeg + bv.z;
    o.w = hv.w * invdeg + bv.w;
    reinterpret_cast<float4*>(out)[i] = o;
}

// ---------------------------------------------------------------------------
// 6) edge scatter: one wave per edge; lanes cover channels {lane, lane+32}.
//    out[t,:] += h[s,:] * dinv[s]*dinv[t]   (f32 atomics, L2-resident)
// ---------------------------------------------------------------------------
__global__ void scatter_kernel(const long long* __restrict__ ei,
                               const float* __restrict__ h,
                               const float* __restrict__ dinv,
                               float* __restrict__ out) {
    int wave = (blockIdx.x * blockDim.x + threadIdx.x) >> 5;
    int lane = threadIdx.x & 31;
    if (wave >= N_EDGES) return;
    int s = (int)ei[wave];
    int t = (int)ei[N_EDGES + wave];
    const float* __restrict__ hs = h + s * D;
    __builtin_prefetch(hs, 0, 0);                 // global_prefetch_b8
    float norm = dinv[s] * dinv[t];
    float* __restrict__ ot = out + t * D;
    float v0 = hs[lane] * norm;
    float v1 = hs[lane + 32] * norm;
    atomicAdd(ot + lane, v0);
    atomicAdd(ot + lane + 32, v1);
}

// ---------------------------------------------------------------------------
// 7) BN stats: per-channel sum / sumsq, two-level reduction
// ---------------------------------------------------------------------------
__global__ void bn_stats_kernel(const float* __restrict__ out,
                                float* __restrict__ bnacc) {  // [0:64)=sum [64:128)=sumsq
    __shared__ float ssum[256];
    __shared__ float ssq[256];
    int tid = blockIdx.x * blockDim.x + threadIdx.x;
    int c = tid & 63;
    int rstride = (gridDim.x * blockDim.x) >> 6;
    float s = 0.0f, q = 0.0f;
    for (int r = tid >> 6; r < N_NODES; r += rstride) {
        float v = out[r * D + c];
        s += v;
        q += v * v;
    }
    ssum[threadIdx.x] = s;
    ssq[threadIdx.x]  = q;
    __syncthreads();
    if (threadIdx.x < 64) {
        float ts = ssum[threadIdx.x] + ssum[threadIdx.x + 64] +
                   ssum[threadIdx.x + 128] + ssum[threadIdx.x + 192];
        float tq = ssq[threadIdx.x] + ssq[threadIdx.x + 64] +
                   ssq[threadIdx.x + 128] + ssq[threadIdx.x + 192];
        atomicAdd(&bnacc[threadIdx.x], ts);
        atomicAdd(&bnacc[D + threadIdx.x], tq);
    }
}

// ---------------------------------------------------------------------------
// 8) BN finalize: scale = gamma*rsqrt(var+eps); shift = beta - mean*scale
// ---------------------------------------------------------------------------
__global__ void bn_finalize_kernel(const float* __restrict__ bnacc,
                                   const float* __restrict__ gamma,
                                   const float* __restrict__ beta,
                                   float* __restrict__ coef) {  // [0:64)=scale [64:128)=shift
    int c = threadIdx.x;
    if (c >= D) return;
    const float invn = 1.0f / (float)N_NODES;
    float mean = bnacc[c] * invn;
    float var  = bnacc[D + c] * invn - mean * mean;
    float sc = gamma[c] * rsqrtf(var + BN_EPS);
    coef[c]     = sc;
    coef[D + c] = beta[c] - mean * sc;
}

// ---------------------------------------------------------------------------
// 9) BN apply + ReLU, float4 vectorized
// ---------------------------------------------------------------------------
__global__ void bn_apply_kernel(float* __restrict__ out, const float* __restrict__ coef) {
    int i = blockIdx.x * blockDim.x + threadIdx.x;  // float4 index
    if (i >= (N_NODES * D) / 4) return;
    int c4 = i & 15;
    float4 sc = reinterpret_cast<const float4*>(coef)[c4];
    float4 sh = reinterpret_cast<const float4*>(coef + D)[c4];
    float4 v = reinterpret_cast<float4*>(out)[i];
    v.x = fmaxf(v.x * sc.x + sh.x, 0.0f);
    v.y = fmaxf(v.y * sc.y + sh.y, 0.0f);
    v.z = fmaxf(v.z * sc.z + sh.z, 0.0f);
    v.w = fmaxf(v.w * sc.w + sh.w, 0.0f);
    reinterpret_cast<float4*>(out)[i] = v;
}

// ---------------------------------------------------------------------------
extern "C" void kernel_launch(void* const* d_in, const int* in_sizes, int n_in,
                              void* d_out, int out_size, void* d_ws, size_t ws_size,
                              hipStream_t stream) {
    const float*     x     = (const float*)d_in[0];
    const long long* ei    = (const long long*)d_in[1];   // int64 per reference
    const float*     W     = (const float*)d_in[2];
    const float*     b     = (const float*)d_in[3];
    const float*     gamma = (const float*)d_in[4];
    const float*     beta  = (const float*)d_in[5];
    float*           out   = (float*)d_out;

    // workspace layout (bytes)
    char* ws = (char*)d_ws;
    float* h     = (float*)(ws);                          // 100000*64*4 = 25,600,000
    float* deg   = (float*)(ws + 25600000);               // 400,000
    float* dinv  = (float*)(ws + 26000000);               // 400,000
    float* bnacc = (float*)(ws + 26400000);               // 128*4
    float* coef  = (float*)(ws + 26400000 + 512);         // 128*4

    const int B256 = 256;

    // 1) init deg=1, bn accumulators=0
    init_kernel<<<(N_NODES + B256 - 1) / B256, B256, 0, stream>>>(deg, bnacc);
    // 2) degree atomics
    degree_kernel<<<(N_EDGES + B256 - 1) / B256, B256, 0, stream>>>(ei, deg);
    // 3) dinv = rsqrt(deg)
    rsqrt_kernel<<<(N_NODES + B256 - 1) / B256, B256, 0, stream>>>(deg, dinv);
    // 4) h = x @ W^T  (WMMA; 6250 node tiles, 8 waves/block)
    gemm_wmma_kernel<<<(N_NODES / 16 + 7) / 8, B256, 0, stream>>>(x, W, h);
    // 5) out = h/deg + bias (self loop)
    seed_out_kernel<<<(N_NODES * D / 4 + B256 - 1) / B256, B256, 0, stream>>>(h, dinv, b, out);
    // 6) edge scatter (one wave per edge)
    scatter_kernel<<<N_EDGES / 8, B256, 0, stream>>>(ei, h, dinv, out);
    // 7) BN statistics
    bn_stats_kernel<<<400, B256, 0, stream>>>(out, bnacc);
    // 8) BN finalize
    bn_finalize_kernel<<<1, 64, 0, stream>>>(bnacc, gamma, beta, coef);
    // 9) BN apply + ReLU
    bn_apply_kernel<<<(N_NODES * D / 4 + B256 - 1) / B256, B256, 0, stream>>>(out, coef);
}